// ModulatedConv2D_74929999446948
// MI455X (gfx1250) — compile-verified
//
#include <hip/hip_runtime.h>
#include <math.h>
#include <stdint.h>

// CDNA5 / gfx1250: wave32, f32 WMMA 16x16x4 + async global->LDS double buffering.
typedef float v2f __attribute__((ext_vector_type(2)));
typedef float v8f __attribute__((ext_vector_type(8)));

#define B_    8
#define H_    64
#define W_    64
#define CIN   512
#define COUT  512
#define KSZ   3

#define TILE_M 128            // spatial positions per block (2 image rows of W=64)
#define TILE_N 64             // output channels per block
#define KC     32             // ci chunk staged per stage
#define LDA    36             // padded LDS stride (floats): banks 4*(9L mod 16) -> conflict-free
#define LDW    36
#define A_STAGE (TILE_M * LDA)            // floats per A buffer (4608)
#define W_STAGE (TILE_N * LDW)            // floats per W buffer (2304)
#define NCHUNK  (CIN / KC)                // 16 ci chunks
#define NTAP    (KSZ * KSZ)               // 9 taps
#define NSTAGE  (NTAP * NCHUNK)           // 144 stages

// ---------------------------------------------------------------------------
// sigma[b,co] = rsqrt(scale^2 * sum_{kh,kw,ci} (kernel*style)^2 + eps)
// ---------------------------------------------------------------------------
__global__ __launch_bounds__(256) void sigma_kernel(const float* __restrict__ kern,
                                                    const float* __restrict__ style,
                                                    float* __restrict__ sigma,
                                                    float scale) {
  int idx = blockIdx.x * blockDim.x + threadIdx.x;   // [0, B*COUT)
  int b  = idx >> 9;
  int co = idx & (COUT - 1);
  float acc = 0.f;
  for (int t = 0; t < NTAP; ++t) {
#pragma unroll 4
    for (int ci = 0; ci < CIN; ++ci) {
      float wv = kern[(t * CIN + ci) * COUT + co];   // coalesced across co
      float sv = style[b * CIN + ci];
      float m  = wv * sv;
      acc = fmaf(m, m, acc);
    }
  }
  sigma[idx] = rsqrtf(scale * scale * acc + 1e-8f);
}

// ---------------------------------------------------------------------------
// Implicit-GEMM modulated conv, double-buffered with async global->LDS copies.
//   out[b,m,co] = (scale*sigma[b,co]) * sum_{tap,ci} x[b,m+tap,ci]*style[b,ci]*kern[tap,ci,co]
// Block: 256 thr = 8 waves (4 M-groups x 2 N-groups), each wave 32x32 of C.
// Stage order: ci-chunk OUTER, tap INNER -> style fragments live in registers
// across the 9 taps of each chunk (no global loads in the steady-state loop).
// ---------------------------------------------------------------------------
__global__ __launch_bounds__(256) void modconv_kernel(const float* __restrict__ x,
                                                      const float* __restrict__ style,
                                                      const float* __restrict__ kern,
                                                      const float* __restrict__ sigma,
                                                      float* __restrict__ out,
                                                      float scale) {
  __shared__ float As[2 * A_STAGE];   // 36864 B
  __shared__ float Ws[2 * W_STAGE];   // 18432 B

  const int b      = blockIdx.z;
  const int mBlock = blockIdx.y * TILE_M;
  const int coBase = blockIdx.x * TILE_N;
  const int tid    = threadIdx.x;
  const int lane   = tid & 31;
  const int lo     = lane & 15;          // fragment row/col within 16
  const int hi     = lane >> 4;          // K-half select
  const int waveId = tid >> 5;
  const int wm     = waveId & 3;         // M-group -> rows wm*32
  const int wn     = waveId >> 2;        // N-group -> cols wn*32
  const int h0     = mBlock >> 6;        // first image row of this block

  // LDS byte offsets (low 32 bits of the flat LDS-aperture address).
  const uint32_t asBase = (uint32_t)(uintptr_t)(&As[0]);
  const uint32_t wsBase = (uint32_t)(uintptr_t)(&Ws[0]);
  const unsigned long long xB = (unsigned long long)(uintptr_t)x;
  const unsigned long long kB = (unsigned long long)(uintptr_t)kern;

  v8f acc[2][2] = {};

  // Issue one stage's loads: A tile via async b128 (zero-fill SAME-pad border),
  // W tile via async b32 transpose-scatter into Ws[n][k].
  // Stage s = cc*NTAP + kk  (cc = ci chunk, kk = tap).
  auto issue_stage = [&](int s, int buf) {
    const int cc     = s / NTAP;
    const int kk     = s - cc * NTAP;    // tap 0..8
    const int ciBase = cc * KC;
    const int kh     = kk / 3 - 1;
    const int kw     = kk % 3 - 1;
    const uint32_t aL = asBase + (uint32_t)(buf * A_STAGE * 4);
    const uint32_t wL = wsBase + (uint32_t)(buf * W_STAGE * 4);
    // A: 128 rows x 32 ci = 1024 float4 -> 4 x b128 per thread
#pragma unroll
    for (int i = 0; i < 4; ++i) {
      int linear = tid + i * 256;
      int m = linear >> 3;               // tile row 0..127
      int q = linear & 7;                // float4 within row
      int h = h0 + (m >> 6) + kh;
      int w = (m & 63) + kw;
      uint32_t lds = aL + (uint32_t)((m * LDA + q * 4) * 4);
      if (h >= 0 && h < H_ && w >= 0 && w < W_) {
        uint32_t go = (uint32_t)((((b * H_ + h) * W_ + w) * CIN + ciBase + q * 4) * 4);
        asm volatile("global_load_async_to_lds_b128 %0, %1, %2"
                     :: "v"(lds), "v"(go), "s"(xB) : "memory");
      } else {
        *(float4*)(&As[buf * A_STAGE + m * LDA + q * 4]) = make_float4(0.f, 0.f, 0.f, 0.f);
      }
    }
    // W: 32 k x 64 n = 2048 scalars -> 8 x b32 per thread, transposed into [n][k]
#pragma unroll
    for (int i = 0; i < 8; ++i) {
      int linear = tid + i * 256;
      int k = linear >> 6;
      int n = linear & 63;
      uint32_t lds = wL + (uint32_t)((n * LDW + k) * 4);
      uint32_t go  = (uint32_t)(((kk * CIN + ciBase + k) * COUT + coBase + n) * 4);
      asm volatile("global_load_async_to_lds_b32 %0, %1, %2"
                   :: "v"(lds), "v"(go), "s"(kB) : "memory");
    }
  };

  issue_stage(0, 0);
  asm volatile("s_wait_asynccnt 0" ::: "memory");
  __syncthreads();

  for (int cc = 0; cc < NCHUNK; ++cc) {
    const int ciBase = cc * KC;
    // Per-lane style fragments for this ci chunk: loaded ONCE, reused for 9 taps.
    v2f sv[KC / 4];
#pragma unroll
    for (int k4 = 0; k4 < KC / 4; ++k4)
      sv[k4] = *(const v2f*)(style + b * CIN + ciBase + k4 * 4 + hi * 2);

    for (int kk = 0; kk < NTAP; ++kk) {
      const int s   = cc * NTAP + kk;
      const int buf = s & 1;
      if (s + 1 < NSTAGE) issue_stage(s + 1, buf ^ 1);   // overlap copy with compute

      const int aOff = buf * A_STAGE;
      const int wOff = buf * W_STAGE;
#pragma unroll
      for (int k4 = 0; k4 < KC / 4; ++k4) {
        v2f a[2], bb[2];
        a[0]  = *(const v2f*)(&As[aOff + (wm * 32 +      lo) * LDA + k4 * 4 + hi * 2]);
        a[1]  = *(const v2f*)(&As[aOff + (wm * 32 + 16 + lo) * LDA + k4 * 4 + hi * 2]);
        a[0] *= sv[k4];
        a[1] *= sv[k4];
        bb[0] = *(const v2f*)(&Ws[wOff + (wn * 32 +      lo) * LDW + k4 * 4 + hi * 2]);
        bb[1] = *(const v2f*)(&Ws[wOff + (wn * 32 + 16 + lo) * LDW + k4 * 4 + hi * 2]);
#pragma unroll
        for (int mt = 0; mt < 2; ++mt)
#pragma unroll
          for (int nt = 0; nt < 2; ++nt)
            acc[mt][nt] = __builtin_amdgcn_wmma_f32_16x16x4_f32(
                false, a[mt], false, bb[nt], (short)0, acc[mt][nt], false, false);
      }

      asm volatile("s_wait_asynccnt 0" ::: "memory");  // stage s+1 landed in LDS
      __syncthreads();
    }
  }

  // ---- epilogue: demodulate and store. C layout: VGPR j -> M=j (lanes 0-15),
  //      M=j+8 (lanes 16-31); N = lane&15.
#pragma unroll
  for (int mt = 0; mt < 2; ++mt) {
#pragma unroll
    for (int nt = 0; nt < 2; ++nt) {
      int   co = coBase + wn * 32 + nt * 16 + lo;
      float s  = scale * sigma[b * COUT + co];
      int mbase = mBlock + wm * 32 + mt * 16 + hi * 8;
#pragma unroll
      for (int j = 0; j < 8; ++j) {
        int gm = mbase + j;
        int h = gm >> 6, w = gm & 63;
        out[((b * H_ + h) * W_ + w) * COUT + co] = acc[mt][nt][j] * s;
      }
    }
  }
}

extern "C" void kernel_launch(void* const* d_in, const int* in_sizes, int n_in,
                              void* d_out, int out_size, void* d_ws, size_t ws_size,
                              hipStream_t stream) {
  const float* x     = (const float*)d_in[0];   // [B,H,W,Cin]
  const float* style = (const float*)d_in[1];   // [B,Cin]
  const float* kern  = (const float*)d_in[2];   // [K,K,Cin,Cout]
  float*       out   = (float*)d_out;           // [B,H,W,Cout]
  float*       sigma = (float*)d_ws;            // [B,Cout] scratch

  const float scale = 1.0f / sqrtf((float)(KSZ * KSZ * CIN));

  sigma_kernel<<<(B_ * COUT) / 256, 256, 0, stream>>>(kern, style, sigma, scale);

  dim3 grid(COUT / TILE_N, (H_ * W_) / TILE_M, B_);
  modconv_kernel<<<grid, 256, 0, stream>>>(x, style, kern, sigma, out, scale);
}